// ThresholdLayer_4449586118868
// MI455X (gfx1250) — compile-verified
//
#include <hip/hip_runtime.h>
#include <hip/hip_bf16.h>

// Masked linear: y = x @ (W*M)^T - bias, M has exactly 32 active cols/row.
// Strategy: compact (cols,vals) per output row, then gather-GEMM with
// V_WMMA_F32_16X16X4_F32 on block-structured B (K = 16 outputs * 32 = 512
// per 16-output group). 8x fewer WMMAs than dense, full f32 precision,
// memory floor ~11us at 23.3 TB/s. x (64MB) stays resident in the 192MB L2.

#define IN_DIM  4096
#define OUT_DIM 4096
#define NACT    32
#define BATCH   4096

// LDS x-block: 16 batch rows, padded row stride so (a) b128 copies stay
// 16B-aligned (4100*4 = 16400 = 0 mod 16) and (b) gather banks
// (4m + col) mod 64 are distinct across the 16 rows of a half-wave.
#define LDS_STRIDE 4100   // floats; 16*4100*4 = 262,400 B (<= 320KB WGP LDS)

typedef __attribute__((ext_vector_type(2))) float v2f;
typedef __attribute__((ext_vector_type(8))) float v8f;

// ---------------------------------------------------------------------------
// Phase 1: per output row, pack indices/values of the 32 active columns in
// ascending column order. One wave32 per row; deterministic. Mask/weight are
// read exactly once -> non-temporal loads.
// ---------------------------------------------------------------------------
__global__ __launch_bounds__(256)
void compact_kernel(const float* __restrict__ w,
                    const int*   __restrict__ mask,
                    int*         __restrict__ colsW,
                    float*       __restrict__ valsW) {
    const int o    = blockIdx.x * 8 + (threadIdx.x >> 5);
    const int lane = threadIdx.x & 31;
    const int*   mrow = mask + (size_t)o * IN_DIM;
    const float* wrow = w    + (size_t)o * IN_DIM;

    const int c0 = lane * (IN_DIM / 32);          // 128 columns per lane
    int cnt = 0;
    for (int i = 0; i < IN_DIM / 32; ++i)
        cnt += (__builtin_nontemporal_load(mrow + c0 + i) != 0) ? 1 : 0;

    // inclusive scan across the wave (wave32), then make it exclusive
    int incl = cnt;
    for (int d = 1; d < 32; d <<= 1) {
        int u = __shfl_up(incl, d, 32);
        if (lane >= d) incl += u;
    }
    int pos = incl - cnt;

    for (int i = 0; i < IN_DIM / 32; ++i) {
        const int c = c0 + i;
        if (__builtin_nontemporal_load(mrow + c) != 0) {
            if (pos < NACT) {                      // row invariant: exactly 32
                colsW[o * NACT + pos] = c;
                valsW[o * NACT + pos] = __builtin_nontemporal_load(wrow + c);
            }
            ++pos;
        }
    }
}

// ---------------------------------------------------------------------------
// Phase 2: one workgroup per (16-row batch block, half of the output groups).
// Stage the x-block to LDS once via async-to-LDS DMA, then 16 waves each own
// 8 output-groups (16 outputs each). Per group: 16x16 tile accumulated by
// 128 x V_WMMA_F32_16X16X4_F32.
//
// WMMA operand layout assumptions (ISA 7.12.2, wave32):
//   A 16x4 f32 (2 VGPR): lanes 0-15 -> M=lane, K={0,1}; lanes 16-31 -> K={2,3}
//   B 4x16  f32 (2 VGPR): lanes hold N=lane&15; v0 = K{0|2}, v1 = K{1|3}
//   C/D 16x16 (8 VGPR): vgpr r -> M = r + 8*(lane>=16), N = lane&15
// ---------------------------------------------------------------------------
__global__ __launch_bounds__(512)
void spmm_wmma_kernel(const float* __restrict__ x,
                      const float* __restrict__ bias,
                      const int*   __restrict__ colsW,
                      const float* __restrict__ valsW,
                      float*       __restrict__ out) {
    __shared__ float ldsX[16 * LDS_STRIDE];

    const int tid    = threadIdx.x;
    const int bBase  = (blockIdx.x >> 1) * 16;    // batch block
    const int ogHalf = blockIdx.x & 1;            // which 128 output groups

    // Force ldsX to escape so the asm "memory" clobbers below are known to
    // potentially write it (otherwise LLVM folds the later LDS reads to undef).
    asm volatile("" :: "v"((void*)ldsX) : "memory");

    // ---- stage x[bBase .. bBase+16, :] into LDS via async DMA (b128) ----
    // ASYNCcnt-tracked: no VGPR round trip, 32 outstanding per wave (< 63).
    // The DMA's LDS operand is the low 32 bits of the flat shared address
    // ptrtoint also marks ldsX as escaping.
    const float* xblk = x + (size_t)bBase * IN_DIM;
    for (int i = tid; i < 16 * (IN_DIM / 4); i += 512) {
        const int m  = i >> 10;                   // IN_DIM/4 = 1024 f4 / row
        const int c4 = (i & 1023) << 2;
        const float* g = xblk + (size_t)m * IN_DIM + c4;
        const unsigned ldsOff = (unsigned)(size_t)(&ldsX[m * LDS_STRIDE + c4]);
        asm volatile("global_load_async_to_lds_b128 %0, %1, off"
                     :: "v"(ldsOff), "v"(g) : "memory");
    }
    asm volatile("s_wait_asynccnt 0x0" ::: "memory");
    __syncthreads();

    const int wave  = tid >> 5;
    const int lane  = tid & 31;
    const int laneN = lane & 15;                  // N index (and A's M index)
    const int h     = lane >> 4;                  // K-half select
    const float* xrow = &ldsX[laneN * LDS_STRIDE];

    // 128 output groups for this block / 16 waves = 8 groups per wave
    for (int ogi = 0; ogi < 8; ++ogi) {
        const int og = ogHalf * 128 + wave + ogi * 16;
        v8f acc = {0.f, 0.f, 0.f, 0.f, 0.f, 0.f, 0.f, 0.f};

        for (int jo = 0; jo < 16; ++jo) {         // output owning this K band
            const int o = og * 16 + jo;
            const int2*   cp = (const int2*)  (colsW + (size_t)o * NACT);
            const float2* vp = (const float2*)(valsW + (size_t)o * NACT);
            const bool mine = (laneN == jo);      // B nonzero only in col jo

            #pragma unroll
            for (int p = 0; p < 8; ++p) {         // K band of 32, chunks of 4
                const int t2 = h + 2 * p;         // pair: t = 2*t2, 2*t2+1
                const int2   cc = cp[t2];
                const float2 vv = vp[t2];
                v2f a, b;
                a.x = xrow[cc.x];                 // gathered x values (LDS)
                a.y = xrow[cc.y];
                b.x = mine ? vv.x : 0.0f;         // block-diagonal B
                b.y = mine ? vv.y : 0.0f;
                acc = __builtin_amdgcn_wmma_f32_16x16x4_f32(
                        false, a, false, b, (short)0, acc, false, false);
            }
        }

        // epilogue: y -= bias; NT stores (write-once, keep x hot in L2)
        const int oBase = og * 16;
        const float bv  = bias[oBase + laneN];
        float* orow = out + (size_t)(bBase + 8 * h) * OUT_DIM + oBase + laneN;
        #pragma unroll
        for (int r = 0; r < 8; ++r)
            __builtin_nontemporal_store(acc[r] - bv, &orow[(size_t)r * OUT_DIM]);
    }
}

// ---------------------------------------------------------------------------
extern "C" void kernel_launch(void* const* d_in, const int* in_sizes, int n_in,
                              void* d_out, int out_size, void* d_ws, size_t ws_size,
                              hipStream_t stream) {
    const float* x    = (const float*)d_in[0];
    const float* w    = (const float*)d_in[1];
    const float* bias = (const float*)d_in[2];
    const int*   mask = (const int*)  d_in[3];
    float*       out  = (float*)d_out;

    int*   colsW = (int*)d_ws;
    float* valsW = (float*)((char*)d_ws + (size_t)OUT_DIM * NACT * sizeof(int));

    compact_kernel<<<OUT_DIM / 8, 256, 0, stream>>>(w, mask, colsW, valsW);
    spmm_wmma_kernel<<<(BATCH / 16) * 2, 512, 0, stream>>>(x, bias, colsW, valsW, out);
}